// HAN_20495583937074
// MI455X (gfx1250) — compile-verified
//
#include <hip/hip_runtime.h>
#include <hip/hip_bf16.h>

typedef __attribute__((ext_vector_type(2))) float v2f;
typedef __attribute__((ext_vector_type(8))) float v8f;

#define WVL_EPS 1e-12f

// ---------------------------------------------------------------------------
// Build the five 16x16 fused-4-level transform matrices (levels 0..19).
// T[p][k][n] = coefficient of input element k in output column n for pass p.
// Output column layout per 16-element group:
//   cols 0..7  : details of level 4p      (8 pairs)
//   cols 8..11 : details of level 4p+1    (4 pairs)
//   cols 12..13: details of level 4p+2    (2 pairs)
//   col  14    : detail  of level 4p+3
//   col  15    : approximation passed to the next pass
// ---------------------------------------------------------------------------
__global__ void HAN_build_T(const float* __restrict__ P, float* __restrict__ Tall) {
    if (threadIdx.x != 0 || blockIdx.x != 0) return;
    float phi0[20], phi1[20];
    for (int l = 0; l < 20; ++l) {
        float p0 = P[2 * l], p1 = P[2 * l + 1];
        float nrm = fmaxf(sqrtf(p0 * p0 + p1 * p1), WVL_EPS);
        phi0[l] = p0 / nrm;
        phi1[l] = p1 / nrm;
    }
    const int cb[4] = {0, 8, 12, 14};
    for (int p = 0; p < 5; ++p) {
        float* T = Tall + p * 256;
        for (int k = 0; k < 16; ++k) {
            float cur[16];
            for (int j = 0; j < 16; ++j) cur[j] = (j == k) ? 1.0f : 0.0f;
            int len = 16;
            for (int s = 0; s < 4; ++s) {
                const int l = 4 * p + s;
                const float q0 = phi1[l], q1 = -phi0[l];
                const int half = len >> 1;
                for (int j = 0; j < half; ++j) {
                    const float x0 = cur[2 * j], x1 = cur[2 * j + 1];
                    T[k * 16 + cb[s] + j] = x0 * q0 + x1 * q1;   // detail
                    cur[j] = x0 * phi0[l] + x1 * phi1[l];        // approx
                }
                len = half;
            }
            T[k * 16 + 15] = cur[0];
        }
    }
}

// ---------------------------------------------------------------------------
// One fused pass of 4 wavelet levels. One wave handles one tile of 256 input
// floats (16 groups x 16 elements) with four chained v_wmma_f32_16x16x4_f32.
//   in       : current approximation signal, length 256*numTiles
//   out      : f_hat (detail bands written at b0, b0/2, b0/4, b0/8)
//   approx   : next-pass approximation, length 16*numTiles
//   b0       : 2^(24 - 4p), base of the level-4p detail band
// ---------------------------------------------------------------------------
__global__ void HAN_wavelet4_pass(const float* __restrict__ in,
                                  float* __restrict__ out,
                                  float* __restrict__ approx,
                                  const float* __restrict__ T,
                                  int numTiles, int b0) {
    const int lane = threadIdx.x & 31;
    const int tile = blockIdx.x * (blockDim.x >> 5) + (threadIdx.x >> 5);
    if (tile >= numTiles) return;                 // wave-uniform: EXEC stays all-1s

    const int n   = lane & 15;                    // B column / A row (m) / D column
    const int khi = (lane >> 4) * 2;              // K sub-offset: 0 (lanes 0-15) or 2

    // B matrix (the 16x16 transform, split into four 4x16 K-chunks).
    // VGPR layout per ISA: v0 = row (4c+khi), v1 = row (4c+khi+1), column = lane&15.
    v2f b[4];
#pragma unroll
    for (int c = 0; c < 4; ++c) {
        const int krow = 4 * c + khi;
        b[c].x = T[krow * 16 + n];
        b[c].y = T[(krow + 1) * 16 + n];
    }

    // A: 16 groups (rows) x 16 elements, chunked by 4 along K.
    const float* tbase = in + 256 * tile;
    v8f acc = {};
#pragma unroll
    for (int c = 0; c < 4; ++c) {
        // element offset = 16*m + 4c + khi  -> even -> 8B aligned float2
        v2f a = *reinterpret_cast<const v2f*>(tbase + 16 * n + 4 * c + khi);
        acc = __builtin_amdgcn_wmma_f32_16x16x4_f32(
            /*neg_a=*/false, a, /*neg_b=*/false, b[c],
            /*c_mod=*/(short)0, acc, /*reuse_a=*/false, /*reuse_b=*/false);
    }

    // Scatter D: column -> destination band, row (= group) -> index in band.
    float* dbase;
    int mult;
    const int b1 = b0 >> 1, b2 = b0 >> 2, b3 = b0 >> 3;
    if (n < 8)       { dbase = out + b0 + n;        mult = 8; }
    else if (n < 12) { dbase = out + b1 + (n - 8);  mult = 4; }
    else if (n < 14) { dbase = out + b2 + (n - 12); mult = 2; }
    else if (n == 14){ dbase = out + b3;            mult = 1; }
    else             { dbase = approx;              mult = 1; }

    const int rowAdd = (lane >> 4) * 8;
    const int g0 = 16 * tile + rowAdd;
#pragma unroll
    for (int r = 0; r < 8; ++r) {
        dbase[mult * (g0 + r)] = acc[r];
    }
}

// ---------------------------------------------------------------------------
// Last 5 levels (20..24) on the 32-element approximation: trivial scalar tail.
// Writes f_hat[0..31].
// ---------------------------------------------------------------------------
__global__ void HAN_tail_levels(const float* __restrict__ fin,
                                const float* __restrict__ P,
                                float* __restrict__ out) {
    if (threadIdx.x != 0 || blockIdx.x != 0) return;
    float cur[32];
    for (int j = 0; j < 32; ++j) cur[j] = fin[j];
    int len = 32;
    for (int l = 20; l < 25; ++l) {
        float p0 = P[2 * l], p1 = P[2 * l + 1];
        const float nrm = fmaxf(sqrtf(p0 * p0 + p1 * p1), WVL_EPS);
        p0 /= nrm; p1 /= nrm;
        const float q0 = p1, q1 = -p0;
        const int half = len >> 1;                // == 2^(24-l) == band base
        for (int j = 0; j < half; ++j) {
            const float x0 = cur[2 * j], x1 = cur[2 * j + 1];
            out[half + j] = x0 * q0 + x1 * q1;
            cur[j] = x0 * p0 + x1 * p1;
        }
        len = half;
    }
    out[0] = cur[0];
}

extern "C" void kernel_launch(void* const* d_in, const int* in_sizes, int n_in,
                              void* d_out, int out_size, void* d_ws, size_t ws_size,
                              hipStream_t stream) {
    (void)in_sizes; (void)n_in; (void)out_size; (void)ws_size;
    const float* f = (const float*)d_in[0];
    const float* P = (const float*)d_in[1];
    float* out = (float*)d_out;

    float* ws = (float*)d_ws;
    float* T  = ws;                    // 5 * 256 floats (pad to 2048)
    float* WA = ws + 2048;             // 2^21 floats (8 MiB)
    float* WB = WA + (1 << 21);        // 2^17 floats (512 KiB)

    HAN_build_T<<<1, 1, 0, stream>>>(P, T);

    struct IO { const float* in; float* ap; };
    const IO io[5] = { {f, WA}, {WA, WB}, {WB, WA}, {WA, WB}, {WB, WA} };

    for (int p = 0; p < 5; ++p) {
        const int numTiles = 1 << (17 - 4 * p);           // 131072, 8192, 512, 32, 2
        const int wavesPerBlock = 8;                      // 256 threads = 8 wave32
        const int blocks = (numTiles + wavesPerBlock - 1) / wavesPerBlock;
        HAN_wavelet4_pass<<<blocks, 256, 0, stream>>>(
            io[p].in, out, io[p].ap, T + p * 256, numTiles, 1 << (24 - 4 * p));
    }

    HAN_tail_levels<<<1, 1, 0, stream>>>(WA, P, out);
}